// HXGATlayer_26903675142557
// MI455X (gfx1250) — compile-verified
//
#include <hip/hip_runtime.h>

#define N_NODES 4096
#define F_DIM   512
#define D_DIM   8
#define HETE    2
#define HMIX    8
#define CHUNK   128
#define NCHUNK  (N_NODES / CHUNK)

typedef float    v2f __attribute__((ext_vector_type(2)));
typedef float    v8f __attribute__((ext_vector_type(8)));
typedef unsigned v4u __attribute__((ext_vector_type(4)));
typedef int      v4i __attribute__((ext_vector_type(4)));
typedef int      v8i __attribute__((ext_vector_type(8)));

#if __has_builtin(__builtin_amdgcn_tensor_load_to_lds)
#define USE_TDM 1
#else
#define USE_TDM 0
#endif

__device__ __forceinline__ float lrelu02(float v) { return v > 0.f ? v : 0.2f * v; }

__device__ __forceinline__ void async_b128(unsigned lds_off, const void* gaddr) {
  // memory -> LDS direct, 16B per lane, tracked by ASYNCcnt
  asm volatile("global_load_async_to_lds_b128 %0, %1, off"
               :: "v"(lds_off), "v"(gaddr)
               : "memory");
}

__device__ __forceinline__ void wait_async0() {
#if __has_builtin(__builtin_amdgcn_s_wait_asynccnt)
  __builtin_amdgcn_s_wait_asynccnt(0);
#else
  asm volatile("s_wait_asynccnt 0x0" ::: "memory");
#endif
}

__device__ __forceinline__ void wait_tensor0() {
#if __has_builtin(__builtin_amdgcn_s_wait_tensorcnt)
  __builtin_amdgcn_s_wait_tensorcnt(0);
#else
  asm volatile("s_wait_tensorcnt 0x0" ::: "memory");
#endif
}

// ---------------- Kernel A: feats = x@W, s_self = feats@a_self, s_neigh = feats@a_neigh
__global__ void __launch_bounds__(256) k_feats(const float* __restrict__ x,
                                               const float* __restrict__ W,
                                               const float* __restrict__ a_self,
                                               const float* __restrict__ a_neigh,
                                               float* __restrict__ feats,
                                               float* __restrict__ s_self,
                                               float* __restrict__ s_neigh) {
  __shared__ float ftile[32][8];
  const int t   = threadIdx.x;
  const int row = blockIdx.x * 32 + (t >> 3);
  const int d   = t & 7;
  const float* xr = x + (size_t)row * F_DIM;
  float acc = 0.f;
#pragma unroll 8
  for (int k = 0; k < F_DIM; ++k) acc = fmaf(xr[k], W[k * D_DIM + d], acc);
  feats[(size_t)row * D_DIM + d] = acc;
  ftile[t >> 3][d] = acc;
  __syncthreads();
  if (t < 32) {
    const int r2 = blockIdx.x * 32 + t;
    float ss = 0.f, sn = 0.f;
#pragma unroll
    for (int dd = 0; dd < 8; ++dd) {
      ss = fmaf(ftile[t][dd], a_self[dd], ss);
      sn = fmaf(ftile[t][dd], a_neigh[dd], sn);
    }
    s_self[r2] = ss;
    s_neigh[r2] = sn;
  }
}

// ---------------- Kernel B: global score upper bound Mg = lrelu(max ss + max sn)
__global__ void __launch_bounds__(512) k_maxscore(const float* __restrict__ s_self,
                                                  const float* __restrict__ s_neigh,
                                                  float* __restrict__ Mg) {
  __shared__ float red[512];
  const int t = threadIdx.x;
  float mss = -3.4e38f, msn = -3.4e38f;
  for (int i = t; i < N_NODES; i += 512) {
    mss = fmaxf(mss, s_self[i]);
    msn = fmaxf(msn, s_neigh[i]);
  }
  red[t] = mss; __syncthreads();
  for (int off = 256; off > 0; off >>= 1) {
    if (t < off) red[t] = fmaxf(red[t], red[t + off]);
    __syncthreads();
  }
  const float MSS = red[0]; __syncthreads();
  red[t] = msn; __syncthreads();
  for (int off = 256; off > 0; off >>= 1) {
    if (t < off) red[t] = fmaxf(red[t], red[t + off]);
    __syncthreads();
  }
  if (t == 0) Mg[0] = lrelu02(MSS + red[0]);
}

// ---------------- Kernel C: fused masked-softmax attention.
// WMMA f32 16x16x4 accumulation; adjacency tiles streamed via the Tensor Data
// Mover (TENSOR_LOAD_TO_LDS, 2D D#, TDM padding reproduces the 132-stride LDS
// layout); feats/s_neigh chunks via async-to-LDS; double-buffered pipeline.
// grid = (N/16, HETE), block = 128 (4 waves).
__global__ void __launch_bounds__(128) k_attn(const int* __restrict__ adjs,
                                              const float* __restrict__ feats,
                                              const float* __restrict__ s_self,
                                              const float* __restrict__ s_neigh,
                                              const float* __restrict__ Mgp,
                                              float* __restrict__ embs) {
  __shared__ __align__(16) int   adj_lds[2][16 * 132];   // padded stride vs bank conflicts
  __shared__ __align__(16) float feats_lds[2][CHUNK * 8];
  __shared__ __align__(16) float sn_lds[2][CHUNK];
  __shared__ float cacc[4][256];
  __shared__ float dacc[4][32];

  const int t    = threadIdx.x;
  const int wave = t >> 5;
  const int lane = t & 31;
  const int M    = lane & 15;   // A row / B,C column index for this lane
  const int half = lane >> 4;   // selects K slots {0,1} or {2,3}
  const int i0   = blockIdx.x * 16;
  const int rel  = blockIdx.y;
  const long long adj_base = (long long)rel * N_NODES * N_NODES;

  const float Mg = Mgp[0];
  const float ss = s_self[i0 + M];

  v8f c = {};           // 16x16 f32 accumulator (cols 8..15 stay zero)
  float dpart = 0.f;    // partial softmax denominator for row M

  // issue one chunk's worth of memory->LDS traffic
  auto issue = [&](int jc, int buf) {
#if USE_TDM
    // ---- adjacency 16x128 tile via TDM (one DMA op, wave 0 only) ----
    if (wave == 0) {
      const unsigned long long ga =
          (unsigned long long)(size_t)(adjs + adj_base + (long long)i0 * N_NODES + jc);
      v4u g0;
      g0[0] = 1u;                                              // count=1 (valid user D#)
      g0[1] = (unsigned)(size_t)&adj_lds[buf][0];              // lds_addr (bytes)
      g0[2] = (unsigned)ga;                                    // global_addr[31:0]
      g0[3] = ((unsigned)(ga >> 32) & 0x01FFFFFFu) | (2u << 30); // addr[56:32] | type=2
      v8i g1;
      g1[0] = (int)((2u << 16)     // data_size = 4 bytes
                  | (1u << 20)     // pad_enable
                  | (6u << 22)     // pad_interval: 128 DWORDs (one tile row)
                  | (3u << 25));   // pad_amount: 4 DWORDs -> LDS row stride 132
      g1[1] = (int)(4096u << 16);  // tensor_dim0[15:0] = 4096  (bits 63:48)
      g1[2] = (int)(4096u << 16);  // tensor_dim1[15:0] = 4096  (bits 95:80)
      g1[3] = (int)(128u  << 16);  // tile_dim0 = 128           (bits 127:112)
      g1[4] = 16;                  // tile_dim1 = 16            (bits 143:128)
      g1[5] = 4096;                // tensor_dim0_stride[31:0]
      g1[6] = 0;
      g1[7] = 0;
      v4i g2 = {};                 // 2D tensor: groups 2/3 unused
      v4i g3 = {};
      v8i g4 = {};                 // extra group (6-arg builtin form): unused
      __builtin_amdgcn_tensor_load_to_lds(g0, g1, g2, g3, g4, 0);
    }
#else
    // ---- fallback: adjacency via 16 async b128 ops (4 per wave) ----
#pragma unroll
    for (int rep = 0; rep < 4; ++rep) {
      const int slot = wave * 128 + rep * 32 + lane;   // 0..511 int4 slots
      const int r = slot >> 5, c4 = slot & 31;
      const int* g = adjs + adj_base + (long long)(i0 + r) * N_NODES + jc + c4 * 4;
      async_b128((unsigned)(size_t)&adj_lds[buf][r * 132 + c4 * 4], g);
    }
#endif
    // feats tile CHUNK x 8 : 8 async b128 ops, 2 per wave
#pragma unroll
    for (int rep = 0; rep < 2; ++rep) {
      const int slot = wave * 64 + rep * 32 + lane;    // 0..255 float4 slots
      const int r = slot >> 1, part = slot & 1;
      const float* g = feats + (size_t)(jc + r) * D_DIM + part * 4;
      async_b128((unsigned)(size_t)&feats_lds[buf][r * 8 + part * 4], g);
    }
    // s_neigh chunk: 512B, one async b128 op by wave 0
    if (wave == 0) {
      const float* g = s_neigh + jc + lane * 4;
      async_b128((unsigned)(size_t)&sn_lds[buf][lane * 4], g);
    }
  };

  issue(0, 0);   // prologue

  for (int ci = 0; ci < NCHUNK; ++ci) {
    const int cur = ci & 1;
    wait_async0();        // my async feats/sn writes into buf `cur` landed
#if USE_TDM
    wait_tensor0();       // wave 0's TDM adjacency tile landed (no-op for waves 1-3)
#endif
    __syncthreads();      // everyone's data visible; prev compute on buf cur^1 done
    if (ci + 1 < NCHUNK) issue((ci + 1) * CHUNK, cur ^ 1);  // overlap with compute

    // wave w covers local columns [w*32, w*32+32): 8 WMMA steps of K=4
#pragma unroll
    for (int s = 0; s < 8; ++s) {
      const int cb = wave * 32 + s * 4;      // local column base of this step
      const int ja = cb + 2 * half;          // this lane's K slots -> columns ja, ja+1
      const int a0i = adj_lds[cur][M * 132 + ja];
      const int a1i = adj_lds[cur][M * 132 + ja + 1];
      const float sc0 = lrelu02(ss + sn_lds[cur][ja]);
      const float sc1 = lrelu02(ss + sn_lds[cur][ja + 1]);
      const float p0 = (a0i != 0) ? __expf(sc0 - Mg) : 0.f;
      const float p1 = (a1i != 0) ? __expf(sc1 - Mg) : 0.f;
      dpart += p0 + p1;
      v2f a; a.x = p0; a.y = p1;                 // A 16x4: lane row M, K = 2*half+{0,1}
      v2f b;                                     // B 4x16: rows ja, ja+1, col M (pad >=8)
      b.x = (M < D_DIM) ? feats_lds[cur][ja * 8 + M] : 0.f;
      b.y = (M < D_DIM) ? feats_lds[cur][(ja + 1) * 8 + M] : 0.f;
      c = __builtin_amdgcn_wmma_f32_16x16x4_f32(false, a, false, b, (short)0, c,
                                                false, false);
    }
  }

  // combine partial C tiles + denominators across the 4 waves
  __syncthreads();
#pragma unroll
  for (int r = 0; r < 8; ++r)
    cacc[wave][(r + 8 * half) * 16 + M] = c[r];  // C layout: VGPR r, lane -> (M=r+8*half, N)
  dacc[wave][lane] = dpart;
  __syncthreads();

  for (int e = t; e < 256; e += 128) {
    const int em = e >> 4, en = e & 15;
    if (en < D_DIM) {
      const float num = cacc[0][e] + cacc[1][e] + cacc[2][e] + cacc[3][e];
      float den = 0.f;
#pragma unroll
      for (int w = 0; w < 4; ++w) den += dacc[w][em] + dacc[w][em + 16];
      const float v = num / den;                 // softmax normalize
      embs[(long long)rel * N_NODES * D_DIM + (size_t)(i0 + em) * D_DIM + en] =
          v > 0.f ? v : 0.f;                     // relu
    }
  }
}

// ---------------- Kernel D: per-row mixer (HETE=2), out = sum_h embs_h * nw_h
__global__ void __launch_bounds__(256) k_mixer(const float* __restrict__ embs,
                                               const float* __restrict__ mk,
                                               const float* __restrict__ maw,
                                               const float* __restrict__ mb,
                                               float* __restrict__ out) {
  const int n = blockIdx.x * blockDim.x + threadIdx.x;
  if (n >= N_NODES) return;
  float e0[8], e1[8];
#pragma unroll
  for (int d = 0; d < 8; ++d) {
    e0[d] = embs[(size_t)n * 8 + d];
    e1[d] = embs[(size_t)N_NODES * 8 + (size_t)n * 8 + d];
  }
  float aw0 = 0.f, aw1 = 0.f;
#pragma unroll
  for (int h = 0; h < HMIX; ++h) {
    float f0 = mb[h], f1 = mb[h];
#pragma unroll
    for (int d = 0; d < 8; ++d) {
      f0 = fmaf(e0[d], mk[d * HMIX + h], f0);
      f1 = fmaf(e1[d], mk[d * HMIX + h], f1);
    }
    aw0 = fmaf(lrelu02(f0), maw[h], aw0);
    aw1 = fmaf(lrelu02(f1), maw[h], aw1);
  }
  const float m  = fmaxf(aw0, aw1);
  float w0 = __expf(aw0 - m), w1 = __expf(aw1 - m);
  const float inv = 1.f / (w0 + w1);
  w0 *= inv; w1 *= inv;
#pragma unroll
  for (int d = 0; d < 8; ++d)
    out[(size_t)n * 8 + d] = e0[d] * w0 + e1[d] * w1;
}

extern "C" void kernel_launch(void* const* d_in, const int* in_sizes, int n_in,
                              void* d_out, int out_size, void* d_ws, size_t ws_size,
                              hipStream_t stream) {
  const float* x       = (const float*)d_in[0];
  const int*   adjs    = (const int*)d_in[1];
  const float* W       = (const float*)d_in[2];
  const float* a_self  = (const float*)d_in[3];
  const float* a_neigh = (const float*)d_in[4];
  const float* mk      = (const float*)d_in[5];
  const float* maw     = (const float*)d_in[6];
  const float* mb      = (const float*)d_in[7];
  float* out = (float*)d_out;

  float* ws      = (float*)d_ws;
  float* feats   = ws;                    // 4096*8   = 32768 floats
  float* s_self  = feats + 32768;         // 4096
  float* s_neigh = s_self + 4096;         // 4096
  float* Mg      = s_neigh + 4096;        // 64 (padded)
  float* embs    = Mg + 64;               // 2*4096*8 = 65536

  k_feats<<<N_NODES / 32, 256, 0, stream>>>(x, W, a_self, a_neigh, feats, s_self, s_neigh);
  k_maxscore<<<1, 512, 0, stream>>>(s_self, s_neigh, Mg);
  dim3 g(N_NODES / 16, HETE);
  k_attn<<<g, 128, 0, stream>>>(adjs, feats, s_self, s_neigh, Mg, embs);
  k_mixer<<<N_NODES / 256, 256, 0, stream>>>(embs, mk, maw, mb, out);
}